// CME_5437428597463
// MI455X (gfx1250) — compile-verified
//
#include <hip/hip_runtime.h>

// CME scan kernel for gfx1250 (MI455X).
// Shapes: fs/alphas [T,B,F]; eta complex[V]; s complex[S,V]; tau[S].
// T=256, B=4, F=64, S=32, V=12.  Chains = B*F*S = 8192.
// Wave layout: 16 chains/wave, 2 lanes per chain (even-v half / odd-v half).
// Eta-projection via V_WMMA_F32_16X16X4_F32 (6 accumulating WMMAs = K 0..23).
// v2: software-pipelined per-t wave-uniform loads + global_prefetch ahead.
// v3: replace 1/z2 IEEE divide (v_div_scale/div_fmas/div_fixup chain) with
//     v_rcp_f32 via __builtin_amdgcn_rcpf -- one trans op instead of ~6 VALU
//     ops + VCC serialization on the scan's critical path.

typedef __attribute__((ext_vector_type(2))) float v2f;
typedef __attribute__((ext_vector_type(8))) float v8f;

#define T_DIM 256
#define BF_DIM 256            // B*F
#define S_DIM 32
#define V_DIM 12
#define R_DIM 8192            // B*F*S chains
#define N_CONST 11.0f         // MAX_FN_EVALS - 1
#define G_CONST 1.0f

__global__ __launch_bounds__(256)
void cme_scan_wmma(const float* __restrict__ fs,
                   const float* __restrict__ alphas,
                   const float* __restrict__ eta,   // complex64 interleaved [V]
                   const float* __restrict__ s,     // complex64 interleaved [S,V]
                   const float* __restrict__ tau,   // [S]
                   float* __restrict__ out)         // til_fs [T,R] then F_last [R,V] cplx
{
    const int lane   = threadIdx.x & 31;
    const int waveId = (blockIdx.x * blockDim.x + threadIdx.x) >> 5;   // 0..511
    const int base   = waveId * 16;        // first chain owned by this wave
    const int half   = lane >> 4;          // 0: even v components, 1: odd v
    const int row    = lane & 15;          // WMMA A-matrix row (chain within wave)
    const int chain  = base + row;         // this lane's chain
    const int bf     = chain >> 5;         // (b*F + f): wave-uniform
    const int sidx   = chain & 31;         // tau-star index

    // Per-lane pole constants and eta weights for its 6 components v = 2j+half.
    // w[2v] = Re(eta_v), w[2v+1] = -Im(eta_v)  (Re(eta*F) = EtaRe*FRe - EtaIm*FIm)
    float sre[6], sim_[6], wre[6], wim[6];
#pragma unroll
    for (int j = 0; j < 6; ++j) {
        const int v = 2 * j + half;
        sre[j]  = s[2 * (sidx * V_DIM + v) + 0];
        sim_[j] = s[2 * (sidx * V_DIM + v) + 1];
        wre[j]  = eta[2 * v + 0];
        wim[j]  = -eta[2 * v + 1];
    }

    // Output scale per chain for the 8 rows this lane may store:
    // til = Re(sum) * (tau/N)^G / tau
    float scl[8];
#pragma unroll
    for (int m = 0; m < 8; ++m) {
        const float tv = tau[(base + half * 8 + m) & 31];
        scl[m] = __powf(tv / N_CONST, G_CONST) / tv;
    }

    float Fre[6], Fim[6];
#pragma unroll
    for (int j = 0; j < 6; ++j) { Fre[j] = 0.0f; Fim[j] = 0.0f; }

    // Software pipeline: values for step t are loaded during step t-1.
    float a_cur  = alphas[bf];
    float f_cur  = fs[bf];

    for (int t = 0; t < T_DIM; ++t) {
        // Issue next iteration's wave-uniform loads now; they retire while the
        // ~40 VALU + 6 WMMA ops below execute (clamped index avoids a branch).
        const int tn = (t + 1 < T_DIM) ? (t + 1) : (T_DIM - 1);
        const float a_nxt = alphas[tn * BF_DIM + bf];
        const float f_nxt = fs[tn * BF_DIM + bf];
        // Warm L2 further ahead (global_prefetch_b8; no counter, no VGPR return)
        const int tp = (t + 8 < T_DIM) ? (t + 8) : (T_DIM - 1);
        __builtin_prefetch(&alphas[tp * BF_DIM + bf], 0, 1);
        __builtin_prefetch(&fs[tp * BF_DIM + bf], 0, 1);

        const float a  = a_cur;
        const float fv = f_cur;

        v8f acc = {};   // 16x16 D tile: every column ends up equal to the eta-dot

#pragma unroll
        for (int j = 0; j < 6; ++j) {
            // z = -a * s_pole
            const float zre = -a * sre[j];
            const float zim = -a * sim_[j];

            // hh = exp(z)
            const float e = __expf(zre);
            float sn, cs;
            __sincosf(zim, &sn, &cs);
            const float hre = e * cs;
            const float him = e * sn;

            // ih = exprel(z): (exp(z)-1)/z, small -> 1 + z/2 (branchless selects,
            // keeps EXEC full for the WMMA below). v_rcp_f32 instead of IEEE div.
            const float z2    = zre * zre + zim * zim;
            const bool  small = z2 < 1e-8f;
            const float inv   = __builtin_amdgcn_rcpf(z2);   // Inf for z2==0, selected away
            const float gre   = hre - 1.0f;
            const float ire_b = (gre * zre + him * zim) * inv;
            const float iim_b = (him * zre - gre * zim) * inv;
            const float ire   = small ? (1.0f + 0.5f * zre) : ire_b;
            const float iim   = small ? (0.5f * zim)        : iim_b;

            // F = F*hh + fv*ih
            const float nre = Fre[j] * hre - Fim[j] * him + fv * ire;
            const float nim = Fre[j] * him + Fim[j] * hre + fv * iim;
            Fre[j] = nre;
            Fim[j] = nim;

            // WMMA j accumulates K = 4j..4j+3:
            //   lanes 0-15  supply (K=4j,4j+1)   = (Re,Im) of v=2j   = this half's state j
            //   lanes 16-31 supply (K=4j+2,4j+3) = (Re,Im) of v=2j+1 = this half's state j
            v2f A;  A.x = nre;     A.y = nim;
            v2f Bm; Bm.x = wre[j]; Bm.y = wim[j];   // broadcast weights across columns
            acc = __builtin_amdgcn_wmma_f32_16x16x4_f32(
                      false, A, false, Bm, (short)0, acc, false, false);
        }

        // D layout: lanes 0-15 hold M=vgpr, lanes 16-31 hold M=vgpr+8; columns equal.
        // Lane 0 stores rows 0..7, lane 16 stores rows 8..15 (two b128 stores).
        if (row == 0) {
            float* o = out + (size_t)t * R_DIM + base + half * 8;
#pragma unroll
            for (int m = 0; m < 8; ++m) o[m] = acc[m] * scl[m];
        }

        a_cur = a_nxt;
        f_cur = f_nxt;
    }

    // F_last: complex64 [B,F,S,V] interleaved, appended after til_fs
    float* fl = out + (size_t)T_DIM * R_DIM;
#pragma unroll
    for (int j = 0; j < 6; ++j) {
        const int v = 2 * j + half;
        fl[2 * (chain * V_DIM + v) + 0] = Fre[j];
        fl[2 * (chain * V_DIM + v) + 1] = Fim[j];
    }
}

extern "C" void kernel_launch(void* const* d_in, const int* in_sizes, int n_in,
                              void* d_out, int out_size, void* d_ws, size_t ws_size,
                              hipStream_t stream) {
    (void)in_sizes; (void)n_in; (void)out_size; (void)d_ws; (void)ws_size;
    const float* fs     = (const float*)d_in[0];
    const float* alphas = (const float*)d_in[1];
    const float* eta    = (const float*)d_in[2];   // complex64 -> interleaved f32
    const float* s      = (const float*)d_in[3];   // complex64 -> interleaved f32
    const float* tau    = (const float*)d_in[4];
    float* out = (float*)d_out;

    // 512 waves total: 16384 threads = 64 blocks x 256 threads (8 waves/block)
    cme_scan_wmma<<<64, 256, 0, stream>>>(fs, alphas, eta, s, tau, out);
}